// DynamicGroup_conv2d_71476845740304
// MI455X (gfx1250) — compile-verified
//
#include <hip/hip_runtime.h>

// Problem constants
#define B_    16
#define CIN_  16
#define OUT_  16
#define H_    320
#define W_    320
#define HW_   (H_ * W_)      // 102400
#define KEXP_ 4
#define HID_  5
#define KTOT_ 144            // CIN*3*3
#define COLS  322            // staged cols: w = -1 .. 320
#define CPAD  18             // channel-minor pad: even (b64 align) + conflict-free

typedef __attribute__((ext_vector_type(2))) float v2f;
typedef __attribute__((ext_vector_type(8))) float v8f;

// ---------------------------------------------------------------------------
// Kernel 1: global average pool.  grid = B*CIN blocks of 256 threads.
// ---------------------------------------------------------------------------
__global__ __launch_bounds__(256) void dyn_pool_kernel(
    const float* __restrict__ x, float* __restrict__ pooled) {
  const int bc = blockIdx.x;                       // b*16 + c
  const float4* p4 = (const float4*)(x + (size_t)bc * HW_);
  float s = 0.0f;
  for (int i = threadIdx.x; i < HW_ / 4; i += 256) {
    float4 v = p4[i];
    s += v.x + v.y + v.z + v.w;
  }
  __shared__ float red[256];
  red[threadIdx.x] = s;
  __syncthreads();
  for (int off = 128; off > 0; off >>= 1) {
    if (threadIdx.x < (unsigned)off) red[threadIdx.x] += red[threadIdx.x + off];
    __syncthreads();
  }
  if (threadIdx.x == 0) pooled[bc] = red[0] * (1.0f / (float)HW_);
}

// ---------------------------------------------------------------------------
// Kernel 2: attention MLP + softmax + expert mix.  1 block, 256 threads,
// thread t = (b = t>>4, j = t&15).
// Writes wmix[b][o][k] with channel-minor K order:  k = (r*3+dx)*16 + c.
// ---------------------------------------------------------------------------
__global__ __launch_bounds__(256) void dyn_attn_kernel(
    const float* __restrict__ pooled, const float* __restrict__ w1,
    const float* __restrict__ w2, const float* __restrict__ b2,
    const float* __restrict__ weight, float* __restrict__ wmix) {
  const int t = threadIdx.x;
  const int b = t >> 4;
  const int j = t & 15;

  // hid = relu(pooled @ w1^T)
  float hid[HID_];
#pragma unroll
  for (int h = 0; h < HID_; ++h) {
    float a = 0.0f;
#pragma unroll
    for (int c = 0; c < CIN_; ++c) a += pooled[b * CIN_ + c] * w1[h * CIN_ + c];
    hid[h] = fmaxf(a, 0.0f);
  }
  // attn column j for k=0..3 : flat index i = k*16 + j
  float av[KEXP_];
#pragma unroll
  for (int k = 0; k < KEXP_; ++k) {
    const int i = k * 16 + j;
    float a = b2[i];
#pragma unroll
    for (int h = 0; h < HID_; ++h) a += hid[h] * w2[i * HID_ + h];
    av[k] = a;
  }
  // softmax over k (axis=1 of [B,4,16])
  float mx = fmaxf(fmaxf(av[0], av[1]), fmaxf(av[2], av[3]));
  float e[KEXP_], sum = 0.0f;
#pragma unroll
  for (int k = 0; k < KEXP_; ++k) { e[k] = expf(av[k] - mx); sum += e[k]; }
  const float inv = 1.0f / sum;

  __shared__ float sm[B_][KEXP_ * OUT_];   // flat layout = k*16 + o'
#pragma unroll
  for (int k = 0; k < KEXP_; ++k) sm[b][k * 16 + j] = e[k] * inv;
  __syncthreads();

  // thread (b, o=j): coeffs via the faithful (k,o)-flat regroup trick
  const int o = j;
  float coef[KEXP_];
#pragma unroll
  for (int k = 0; k < KEXP_; ++k) coef[k] = sm[b][o * KEXP_ + k];

  // wmix[b][o][(r*3+dx)*16 + c] = sum_e coef[e] * weight[o][c][r][dx][e]
  const float* wsrc = weight + (size_t)o * KTOT_ * KEXP_;  // [c][r][dx][e]
  float* wdst = wmix + ((size_t)b * OUT_ + o) * KTOT_;
  for (int c = 0; c < CIN_; ++c) {
#pragma unroll
    for (int rd = 0; rd < 9; ++rd) {
      float acc = 0.0f;
#pragma unroll
      for (int e2 = 0; e2 < KEXP_; ++e2)
        acc += coef[e2] * wsrc[(c * 9 + rd) * KEXP_ + e2];
      wdst[rd * 16 + c] = acc;
    }
  }
}

// ---------------------------------------------------------------------------
// Kernel 3: per-sample 3x3 conv as implicit GEMM on V_WMMA_F32_16X16X4_F32.
// grid = B * (H/2) blocks; block = 256 threads (8 waves).
// LDS staged channel-minor:  xs[(row*COLS + col)*CPAD + c]
//   -> each lane's B pair (c, c+1) is one aligned ds_load_b64 with a
//      compile-time immediate off a single per-tile base register, landing
//      directly in the even VGPR pair WMMA wants (no register shuffles),
//      and wave32 b64 loads touch all 64 LDS banks exactly once.
// ---------------------------------------------------------------------------
__global__ __launch_bounds__(256) void dyn_conv_kernel(
    const float* __restrict__ x, const float* __restrict__ wmix,
    float* __restrict__ out) {
  const int blk = blockIdx.x;
  const int b = blk / (H_ / 2);
  const int h0 = (blk % (H_ / 2)) * 2;

  __shared__ float xs[4 * COLS * CPAD];   // rows h0-1..h0+2, 92736 bytes

  // ---- stage x rows h0-1 .. h0+2 (zero-padded halo), transposed ----
  const float* xb = x + (size_t)b * CIN_ * HW_;
  for (int i = threadIdx.x; i < CIN_ * 4 * COLS; i += 256) {
    const int col = i % COLS;          // col = w + 1
    const int cr = i / COLS;           // c*4 + r
    const int r = cr & 3;
    const int c = cr >> 2;
    const int hh = h0 - 1 + r;
    const int ww = col - 1;
    float v = 0.0f;
    if (hh >= 0 && hh < H_ && ww >= 0 && ww < W_)
      v = xb[c * HW_ + hh * W_ + ww];
    xs[(r * COLS + col) * CPAD + c] = v;
  }
  __syncthreads();

  const int lane = threadIdx.x & 31;
  const int wave = threadIdx.x >> 5;   // 0..7
  const int m = lane & 15;             // A row (out ch) / B col (pixel)
  const int hi = lane >> 4;            // 0|1 : K half per the 16x4 f32 layout

  // ---- A fragments: wmix[b][m][4s + 2*hi + {0,1}] ----
  const float* wb = wmix + ((size_t)b * OUT_ + m) * KTOT_;
  v2f a[36];
#pragma unroll
  for (int s = 0; s < 36; ++s) {
    a[s].x = wb[4 * s + 2 * hi + 0];
    a[s].y = wb[4 * s + 2 * hi + 1];
  }

  // ---- 5 pixel-tiles per wave (2 rows x 20 tiles = 40 per block) ----
#pragma unroll
  for (int it = 0; it < 5; ++it) {
    const int tt = wave * 5 + it;        // 0..39
    const int lr = tt / 20;              // local output row 0|1
    const int wbase = (tt % 20) * 16;    // pixel base
    // per-tile base (dwords); even => every b64 below is 8B-aligned
    const int base = (lr * COLS + wbase + m) * CPAD + 2 * hi;

    v8f acc = {};
#pragma unroll
    for (int s = 0; s < 36; ++s) {
      const int cb = (4 * s) & 15;       // compile-time channel base (even)
      const int rd = (4 * s) >> 4;       // compile-time (r,dx) group
      const int r  = rd / 3;
      const int dx = rd % 3;
      const v2f bf = *(const v2f*)(xs + base + (r * COLS + dx) * CPAD + cb);
      acc = __builtin_amdgcn_wmma_f32_16x16x4_f32(
          false, a[s], false, bf, (short)0, acc, false, false);
    }

    // D layout: lane(m,hi), VGPR j -> D[o = j + 8*hi][n = m]
    float* ob = out + (size_t)b * OUT_ * HW_ + (h0 + lr) * W_ + wbase + m;
#pragma unroll
    for (int jj = 0; jj < 8; ++jj)
      ob[(size_t)(jj + 8 * hi) * HW_] = acc[jj];
  }
}

// ---------------------------------------------------------------------------
extern "C" void kernel_launch(void* const* d_in, const int* in_sizes, int n_in,
                              void* d_out, int out_size, void* d_ws, size_t ws_size,
                              hipStream_t stream) {
  (void)in_sizes; (void)n_in; (void)out_size; (void)ws_size;
  const float* x      = (const float*)d_in[0];
  const float* w1     = (const float*)d_in[1];
  const float* w2     = (const float*)d_in[2];
  const float* b2     = (const float*)d_in[3];
  const float* weight = (const float*)d_in[4];
  float* out = (float*)d_out;

  float* pooled = (float*)d_ws;                       // 256 floats
  float* wmix   = (float*)d_ws + 256;                 // 16*16*144 floats

  dyn_pool_kernel<<<B_ * CIN_, 256, 0, stream>>>(x, pooled);
  dyn_attn_kernel<<<1, 256, 0, stream>>>(pooled, w1, w2, b2, weight, wmix);
  dyn_conv_kernel<<<B_ * (H_ / 2), 256, 0, stream>>>(x, wmix, out);
}